// STSEarlyFusionTimeMatch_70643622085284
// MI455X (gfx1250) — compile-verified
//
#include <hip/hip_runtime.h>

// ---------------------------------------------------------------------------
// Stereo cost-volume construction (masked broadcast + disparity shift-gather).
//
// Pure data movement: 503 MB of f32 output, 8 MB of inputs (L2-resident).
// Store-bound: ~503MB / 23.3 TB/s ~= 22 us floor on MI455X. No matrix math
// exists in this op, so WMMA is deliberately NOT used (its C/D lane layout
// would scatter stores across (c,d) and destroy coalescing). CDNA5-specific
// paths used instead:
//   * global_load_async_to_lds_b128 + s_wait_asynccnt to stage each input
//     row in LDS (keeps the VMEM pipe free for the store stream)
//   * b128 non-temporal stores (output > L2, never re-read -> stream it)
// ---------------------------------------------------------------------------

typedef float v4f __attribute__((ext_vector_type(4)));
typedef int   v4i __attribute__((ext_vector_type(4)));

#define CV_F 32      // channels per input tensor
#define CV_H 96
#define CV_W 160

#if defined(__AMDGCN__)
typedef __attribute__((address_space(1))) v4i g_v4i;   // global (AS1)
typedef __attribute__((address_space(3))) v4i l_v4i;   // LDS    (AS3)
#endif

__global__ __launch_bounds__(256)
void cost_volume_kernel(const float* __restrict__ l0,
                        const float* __restrict__ r0,
                        const float* __restrict__ l1,
                        const float* __restrict__ r1,
                        const int*  __restrict__ max_disp_p,
                        float* __restrict__ out)
{
    const int D = max_disp_p[0] / 3;   // 64 for max_disp = 192

    __shared__ float row[CV_W];        // one input row, 640 B

    const int blk  = blockIdx.x;       // 0 .. 128*96-1
    const int h    = blk % CV_H;
    const int c    = blk / CV_H;       // output channel 0..127
    const int half = c >> 5;           // 0:l0 1:r0 2:l1 3:r1 (concat order)
    const int cc   = c & (CV_F - 1);
    const int isR  = half & 1;         // right tensors are shifted by d

    const float* src  = (half == 0) ? l0 : (half == 1) ? r0
                      : (half == 2) ? l1 : r1;
    const float* srow = src + ((size_t)cc * CV_H + h) * CV_W;

    const int tid = threadIdx.x;

    // ---- stage the input row into LDS ------------------------------------
#if defined(__AMDGCN__) && __has_builtin(__builtin_amdgcn_global_load_async_to_lds_b128)
    if (tid < CV_W / 4) {
        // 40 lanes x 16B = 640B async DMA into LDS (ASYNCcnt-tracked).
        __builtin_amdgcn_global_load_async_to_lds_b128(
            (g_v4i*)(srow + tid * 4),
            (l_v4i*)(&row[tid * 4]),
            /*offset=*/0, /*cpol=*/0);
    }
  #if __has_builtin(__builtin_amdgcn_s_wait_asynccnt)
    __builtin_amdgcn_s_wait_asynccnt(0);
  #else
    asm volatile("s_wait_asynccnt 0" ::: "memory");
  #endif
    __syncthreads();
#else
    if (tid < CV_W) row[tid] = srow[tid];
    __syncthreads();
#endif

    // ---- write D disparity planes of this (c,h) row ----------------------
    // out[((c*D + d)*H + h)*W + w]
    float* obase = out + (size_t)c * D * (CV_H * CV_W) + (size_t)h * CV_W;

    const int nq = D * (CV_W / 4);         // float4 stores for this row block
    for (int idx = tid; idx < nq; idx += 256) {
        const int w4 = idx % (CV_W / 4);   // lane-consecutive -> coalesced
        const int d  = idx / (CV_W / 4);
        const int w  = w4 * 4;

        v4f v;
#pragma unroll
        for (int j = 0; j < 4; ++j) {
            const int ww = w + j;
            int ri = ww - (isR ? d : 0);   // right: shifted gather
            ri = ri < 0 ? 0 : ri;          // clamp (masked lanes read [0])
            const float t = row[ri];       // ds_load, LDS pipe
            v[j] = (ww >= d) ? t : 0.0f;   // mask == multiply by {0,1}
        }
        __builtin_nontemporal_store(
            v, (v4f*)(obase + (size_t)d * (CV_H * CV_W) + w));
    }
}

extern "C" void kernel_launch(void* const* d_in, const int* in_sizes, int n_in,
                              void* d_out, int out_size, void* d_ws, size_t ws_size,
                              hipStream_t stream) {
    (void)in_sizes; (void)n_in; (void)out_size; (void)d_ws; (void)ws_size;

    const float* l0 = (const float*)d_in[0];
    const float* r0 = (const float*)d_in[1];
    const float* l1 = (const float*)d_in[2];
    const float* r1 = (const float*)d_in[3];
    const int*   md = (const int*)d_in[4];
    float* out = (float*)d_out;

    dim3 grid(4 * CV_F * CV_H);   // one block per (output channel, row)
    dim3 block(256);
    hipLaunchKernelGGL(cost_volume_kernel, grid, block, 0, stream,
                       l0, r0, l1, r1, md, out);
}